// HomologyPressure_76441827934279
// MI455X (gfx1250) — compile-verified
//
#include <hip/hip_runtime.h>

#define BLOCK 256
#define PER_THREAD 16
#define TILE (BLOCK * PER_THREAD)  // 4096 elements per block-tile

// ---------------- feature probes (compile-safe) ----------------
#if defined(__has_builtin)
#  if __has_builtin(__builtin_amdgcn_global_load_async_to_lds_b32) && \
      __has_builtin(__builtin_amdgcn_s_wait_asynccnt)
#    define HAVE_ASYNC_LDS 1
#  endif
#  if __has_builtin(__builtin_amdgcn_wmma_f32_16x16x4_f32)
#    define HAVE_WMMA_F32X4 1
#  endif
#endif
#ifndef HAVE_ASYNC_LDS
#  define HAVE_ASYNC_LDS 0
#endif
#ifndef HAVE_WMMA_F32X4
#  define HAVE_WMMA_F32X4 0
#endif

typedef __attribute__((ext_vector_type(2))) float v2f;
typedef __attribute__((ext_vector_type(8))) float v8f;

// ---------------- ordered-float <-> uint encoding (monotonic) ----------------
__device__ __forceinline__ unsigned fenc(float f) {
  unsigned u = __float_as_uint(f);
  return (u & 0x80000000u) ? ~u : (u | 0x80000000u);
}
__device__ __forceinline__ float fdec(unsigned u) {
  unsigned v = (u & 0x80000000u) ? (u & 0x7FFFFFFFu) : ~u;
  return __uint_as_float(v);
}

__device__ __forceinline__ void flushSeg(int seg, float cnt, float isum,
                                         float pmx, float pmn, float cmn,
                                         float* cnt_arr, float* isum_arr,
                                         unsigned* pmax_arr, unsigned* pmin_arr,
                                         unsigned* cmin_arr) {
  atomicAdd(cnt_arr + seg, cnt);
  atomicAdd(isum_arr + seg, isum);
  atomicMax(pmax_arr + seg, fenc(pmx));
  atomicMin(pmin_arr + seg, fenc(pmn));
  atomicMin(cmin_arr + seg, fenc(cmn));
}

// ---------------- init: identities for the segment aggregates ----------------
__global__ __launch_bounds__(BLOCK) void hp_init(float* cnt_arr, float* isum_arr,
                                                 unsigned* pmax_arr, unsigned* pmin_arr,
                                                 unsigned* cmin_arr,
                                                 const int* __restrict__ ncyc_ptr,
                                                 float* out) {
  int n = *ncyc_ptr;
  int stride = gridDim.x * blockDim.x;
  for (int i = blockIdx.x * blockDim.x + threadIdx.x; i < n; i += stride) {
    cnt_arr[i]  = 0.0f;
    isum_arr[i] = 0.0f;
    pmax_arr[i] = 0u;           // <= fenc(-inf): identity for ordered max
    pmin_arr[i] = 0xFFFFFFFFu;  // >= fenc(+inf): identity for ordered min
    cmin_arr[i] = 0xFFFFFFFFu;
  }
  if (blockIdx.x == 0 && threadIdx.x == 0) out[0] = 0.0f;
}

// ---------------- pass 1: gather + sigmoid + sorted segmented aggregates ----
__global__ __launch_bounds__(BLOCK) void hp_pass1(
    const float* __restrict__ pressures, const float* __restrict__ coh,
    const int* __restrict__ node_idx, const int* __restrict__ seg_ids, int T,
    float* __restrict__ cnt_arr, float* __restrict__ isum_arr,
    unsigned* __restrict__ pmax_arr, unsigned* __restrict__ pmin_arr,
    unsigned* __restrict__ cmin_arr) {
  __shared__ int s_nid[TILE];
  __shared__ int s_sid[TILE];

  for (long long tile0 = (long long)blockIdx.x * TILE; tile0 < T;
       tile0 += (long long)gridDim.x * TILE) {
    int remaining = T - (int)tile0;
    if (remaining > TILE) remaining = TILE;

#if HAVE_ASYNC_LDS
    // builtin signature (from round-1 diagnostic): param0 is non-const
    // addrspace(1) int*, param1 addrspace(3) int*, then offset, cpol.
    typedef __attribute__((address_space(1))) int* g_ptr;
    typedef __attribute__((address_space(3))) int* ld_ptr;
    if (remaining == TILE) {
      for (int i = threadIdx.x; i < TILE; i += BLOCK) {
        __builtin_amdgcn_global_load_async_to_lds_b32(
            (g_ptr)(node_idx + tile0 + i), (ld_ptr)(s_nid + i), 0, 0);
        __builtin_amdgcn_global_load_async_to_lds_b32(
            (g_ptr)(seg_ids + tile0 + i), (ld_ptr)(s_sid + i), 0, 0);
      }
      __builtin_amdgcn_s_wait_asynccnt(0);
    } else
#endif
    {
      for (int i = threadIdx.x; i < TILE; i += BLOCK) {
        int g = (int)tile0 + i;
        // speculative prefetch of the next grid-stride tile (gfx1250: global_prefetch_b8)
        if (g + (int)gridDim.x * TILE < T)
          __builtin_prefetch(node_idx + g + (int)gridDim.x * TILE, 0, 0);
        s_nid[i] = (i < remaining) ? node_idx[g] : 0;
        s_sid[i] = (i < remaining) ? seg_ids[g] : -1;
      }
    }
    __syncthreads();

    int base = threadIdx.x * PER_THREAD;
    int seg = -1;
    float cnt = 0.0f, isum = 0.0f, pmx = 0.0f, pmn = 0.0f, cmn = 0.0f;
#pragma unroll
    for (int j = 0; j < PER_THREAD; ++j) {
      int i = base + j;
      if (i < remaining) {
        int sid = s_sid[i];
        int nid = s_nid[i];
        float p = pressures[nid];   // L2-resident gather (8MB table)
        float c = coh[nid];
        float ind = 1.0f / (1.0f + __expf(-10.0f * (p - 0.5f)));
        if (sid != seg) {
          if (seg >= 0)
            flushSeg(seg, cnt, isum, pmx, pmn, cmn,
                     cnt_arr, isum_arr, pmax_arr, pmin_arr, cmin_arr);
          seg = sid;
          cnt = 0.0f; isum = 0.0f; pmx = p; pmn = p; cmn = c;
        }
        cnt += 1.0f;
        isum += ind;
        pmx = fmaxf(pmx, p);
        pmn = fminf(pmn, p);
        cmn = fminf(cmn, c);
      }
    }
    if (seg >= 0)
      flushSeg(seg, cnt, isum, pmx, pmn, cmn,
               cnt_arr, isum_arr, pmax_arr, pmin_arr, cmin_arr);
    __syncthreads();
  }
}

// ---------------- pass 2: per-cycle nonlinearity + global sum ----------------
__global__ __launch_bounds__(BLOCK) void hp_pass2(
    const float* __restrict__ cnt_arr, const float* __restrict__ isum_arr,
    const unsigned* __restrict__ pmax_arr, const unsigned* __restrict__ pmin_arr,
    const unsigned* __restrict__ cmin_arr, const int* __restrict__ ncyc_ptr,
    float* out) {
  int n = *ncyc_ptr;
  int stride = gridDim.x * blockDim.x;
  float partial = 0.0f;
  for (int i = blockIdx.x * blockDim.x + threadIdx.x; i < n; i += stride) {
    float cnt = cnt_arr[i];
    if (cnt > 0.0f) {
      float pers = fdec(pmax_arr[i]) - fdec(pmin_arr[i]);
      float cm = fdec(cmin_arr[i]);
      partial += isum_arr[i] * __fsqrt_rn(cnt) * (1.0f + 2.0f * pers) *
                 (1.0f - 0.5f * cm);
    }
  }

  // Wave32 reduction. WMMA ones-trick: A = 16x4 ones, B carries the 32 lane
  // partials (2 slots/lane, second zero) -> D[m][n] = sum_k B[k][n] for all m.
  // Summing D VGPR0 over all 32 lanes = row0-sum + row8-sum = 2 * total (exact f32).
#if HAVE_WMMA_F32X4
  v2f a; a[0] = 1.0f; a[1] = 1.0f;
  v2f b; b[0] = partial; b[1] = 0.0f;
  v8f cacc = {};
  v8f dmat = __builtin_amdgcn_wmma_f32_16x16x4_f32(false, a, false, b,
                                                   (short)0, cacc, false, false);
  float w = dmat[0];
  for (int off = 16; off > 0; off >>= 1) w += __shfl_xor(w, off, 32);
  w *= 0.5f;
#else
  float w = partial;
  for (int off = 16; off > 0; off >>= 1) w += __shfl_xor(w, off, 32);
#endif
  if ((threadIdx.x & 31) == 0) atomicAdd(out, w);
}

// ---------------- host entry ----------------
extern "C" void kernel_launch(void* const* d_in, const int* in_sizes, int n_in,
                              void* d_out, int out_size, void* d_ws, size_t ws_size,
                              hipStream_t stream) {
  const float* pressures = (const float*)d_in[0];
  const float* coh       = (const float*)d_in[1];
  const int*   node_idx  = (const int*)d_in[2];
  const int*   seg_ids   = (const int*)d_in[3];
  const int*   ncyc_ptr  = (const int*)d_in[4];  // device-resident scalar
  int T = in_sizes[2];
  float* out = (float*)d_out;

  // carve workspace: 5 arrays of up to `cap` 4-byte aggregates
  size_t cap = ws_size / (5 * sizeof(float));
  float*    cnt_arr  = (float*)d_ws;
  float*    isum_arr = cnt_arr + cap;
  unsigned* pmax_arr = (unsigned*)(isum_arr + cap);
  unsigned* pmin_arr = pmax_arr + cap;
  unsigned* cmin_arr = pmin_arr + cap;

  hp_init<<<2048, BLOCK, 0, stream>>>(cnt_arr, isum_arr, pmax_arr, pmin_arr,
                                      cmin_arr, ncyc_ptr, out);

  int tiles = (T + TILE - 1) / TILE;
  if (tiles < 1) tiles = 1;
  hp_pass1<<<tiles, BLOCK, 0, stream>>>(pressures, coh, node_idx, seg_ids, T,
                                        cnt_arr, isum_arr, pmax_arr, pmin_arr,
                                        cmin_arr);

  hp_pass2<<<2048, BLOCK, 0, stream>>>(cnt_arr, isum_arr, pmax_arr, pmin_arr,
                                       cmin_arr, ncyc_ptr, out);
}